// DynamicEdgeConstruction_55834574848108
// MI455X (gfx1250) — compile-verified
//
#include <hip/hip_runtime.h>

typedef __attribute__((ext_vector_type(2))) float v2f;
typedef __attribute__((ext_vector_type(8))) float v8f;

#define BB   4
#define NN   4096
#define CC   256
#define DK   64
#define TOPK 8
#define LDSROW 68                 // padded K-tile row stride (floats): conflict-free b64 reads
#define CTILES ((NN / 4) / 16)    // 64 column tiles per wave strip

// ---------------------------------------------------------------------------
// Kernel 1: Q = (x @ Wq) * DK^-0.5 ; K = x @ Wk
// One wave per 16-row tile. f32 WMMA 16x16x4, K-loop over C in steps of 4.
// ---------------------------------------------------------------------------
__global__ __launch_bounds__(32)
void qk_proj_kernel(const float* __restrict__ x,
                    const float* __restrict__ Wq,
                    const float* __restrict__ Wk,
                    float* __restrict__ Q,
                    float* __restrict__ K) {
  const int rbase = blockIdx.x * 16;        // row tile within flattened [B*N]
  const int lane  = threadIdx.x;
  const int l15   = lane & 15;
  const int half  = lane >> 4;

  v8f accq[4] = {{0.f}}; v8f acck[4] = {{0.f}};

  for (int kk = 0; kk < CC / 4; ++kk) {
    const int kb = kk * 4 + 2 * half;       // this lane's K pair {kb, kb+1}
    v2f a = *(const v2f*)(x + (size_t)(rbase + l15) * CC + kb);
#pragma unroll
    for (int nt = 0; nt < 4; ++nt) {
      const int n = nt * 16 + l15;
      v2f bq; bq.x = Wq[(kb + 0) * DK + n]; bq.y = Wq[(kb + 1) * DK + n];
      v2f bk; bk.x = Wk[(kb + 0) * DK + n]; bk.y = Wk[(kb + 1) * DK + n];
      accq[nt] = __builtin_amdgcn_wmma_f32_16x16x4_f32(false, a, false, bq,
                                                       (short)0, accq[nt], false, false);
      acck[nt] = __builtin_amdgcn_wmma_f32_16x16x4_f32(false, a, false, bk,
                                                       (short)0, acck[nt], false, false);
    }
  }

  const float scale = 0.125f;               // DK^-0.5, folded into Q
#pragma unroll
  for (int nt = 0; nt < 4; ++nt) {
#pragma unroll
    for (int i = 0; i < 8; ++i) {
      const int row = rbase + i + 8 * half;
      const int col = nt * 16 + l15;
      Q[(size_t)row * DK + col] = accq[nt][i] * scale;
      K[(size_t)row * DK + col] = acck[nt][i];
    }
  }
}

// ---------------------------------------------------------------------------
// Kernel 2: fused S = Q K^T (f32 WMMA) + streaming exact top-8 + softmax +
// scatter. 4 waves / block, wave-private column strips. K tiles are staged
// LDS-side via GLOBAL_LOAD_ASYNC_TO_LDS_B128 (ASYNCcnt) with double buffering
// so the copy of tile t+1 overlaps the 16-WMMA chain of tile t.
// ---------------------------------------------------------------------------
__global__ __launch_bounds__(128)
void topk_attn_kernel(const float* __restrict__ Q,
                      const float* __restrict__ K,
                      float* __restrict__ out) {
  __shared__ float kbuf[4][2][16 * LDSROW]; // 4 waves x double buffer (4.25KB each)
  __shared__ float thr[4][16];              // per-wave per-row 8th-best value
  __shared__ float listV[4][16][8];         // per-wave per-row sorted top-8 values
  __shared__ int   listI[4][16][8];
  __shared__ int   candN[4][16];
  __shared__ float candV[4][16][16];
  __shared__ int   candI[4][16][16];

  const int tile  = blockIdx.x;             // over B*N/16
  const int b     = tile / (NN / 16);
  const int rtile = (tile % (NN / 16)) * 16;
  const int tid   = threadIdx.x;
  const int w     = tid >> 5;
  const int lane  = tid & 31;
  const int l15   = lane & 15;
  const int half  = lane >> 4;

  if (lane < 16) {
    thr[w][lane] = -__builtin_inff();
#pragma unroll
    for (int e = 0; e < TOPK; ++e) {
      listV[w][lane][e] = -__builtin_inff();
      listI[w][lane][e] = 0;
    }
  }
  __builtin_amdgcn_wave_barrier();

  // Preload this block's 16 Q (A-operand) fragments: 16 x float2 per lane.
  const float* Qb = Q + ((size_t)b * NN + rtile) * DK;
  v2f aq[16];
#pragma unroll
  for (int kk = 0; kk < 16; ++kk)
    aq[kk] = *(const v2f*)(Qb + (size_t)l15 * DK + kk * 4 + 2 * half);

  const float* Kb = K + (size_t)b * NN * DK;
  const int    cw = w * (NN / 4);           // wave's column strip base

  // This lane's 16B chunk mapping for the async tile copy (8 x B128 per wave,
  // 2 source rows per issue; per-lane padded LDS destination).
  const int ch_row0  = lane >> 4;           // row advanced by 2 per issue
  const int ch_col4  = (lane & 15) * 4;     // float column of this 16B chunk

  // Kick off the async copy of tile 0 into buffer 0.
  {
    const float* src = Kb + (size_t)cw * DK;
#pragma unroll
    for (int i = 0; i < 8; ++i) {
      const int row = ch_row0 + i * 2;
      unsigned ldst = (unsigned)(uintptr_t)&kbuf[w][0][row * LDSROW + ch_col4];
      const float* g = src + (row * DK + ch_col4);
      asm volatile("global_load_async_to_lds_b128 %0, %1, off"
                   :: "v"(ldst), "v"(g) : "memory");
    }
  }

  for (int t = 0; t < CTILES; ++t) {
    const int ctile = cw + t * 16;

    // Issue the copy of tile t+1 into the other buffer, then wait for tile t.
    if (t + 1 < CTILES) {
      const float* src = Kb + (size_t)(ctile + 16) * DK;
      const int bsel = (t + 1) & 1;
#pragma unroll
      for (int i = 0; i < 8; ++i) {
        const int row = ch_row0 + i * 2;
        unsigned ldst = (unsigned)(uintptr_t)&kbuf[w][bsel][row * LDSROW + ch_col4];
        const float* g = src + (row * DK + ch_col4);
        asm volatile("global_load_async_to_lds_b128 %0, %1, off"
                     :: "v"(ldst), "v"(g) : "memory");
      }
      asm volatile("s_wait_asynccnt 0x8" ::: "memory");  // oldest 8 (tile t) done
    } else {
      asm volatile("s_wait_asynccnt 0x0" ::: "memory");
    }

    // --- 16x16 score tile: 16 chained f32 WMMAs over K-dim 64 ---
    const float* kb0 = &kbuf[w][t & 1][0];
    v8f acc = {0.f};
#pragma unroll
    for (int kk = 0; kk < 16; ++kk) {
      v2f bk = *(const v2f*)(kb0 + l15 * LDSROW + kk * 4 + 2 * half);
      acc = __builtin_amdgcn_wmma_f32_16x16x4_f32(false, aq[kk], false, bk,
                                                  (short)0, acc, false, false);
    }

    // --- streaming top-8 maintenance (wave-private) ---
    if (lane < 16) candN[w][lane] = 0;
    __builtin_amdgcn_wave_barrier();

#pragma unroll
    for (int i = 0; i < 8; ++i) {
      const int   r = i + 8 * half;
      const float v = acc[i];
      if (v > thr[w][r]) {                  // LDS broadcast read
        const int p = atomicAdd(&candN[w][r], 1);
        candV[w][r][p] = v;
        candI[w][r][p] = ctile + l15;
      }
    }
    __builtin_amdgcn_wave_barrier();

    if (lane < 16) {                        // serial sorted-insert per row
      const int r = lane;
      const int n = candN[w][r];
      for (int c = 0; c < n; ++c) {
        float v   = candV[w][r][c];
        int   idx = candI[w][r][c];
        if (v > listV[w][r][TOPK - 1]) {
#pragma unroll
          for (int e = 0; e < TOPK; ++e) {
            if (v > listV[w][r][e]) {
              const float tv = listV[w][r][e]; const int ti = listI[w][r][e];
              listV[w][r][e] = v;  listI[w][r][e] = idx;
              v = tv; idx = ti;
            }
          }
        }
      }
      thr[w][r] = listV[w][r][TOPK - 1];
    }
    __builtin_amdgcn_wave_barrier();
  }

  __syncthreads();

  // --- final 4-way merge per row, softmax over 8, scatter ---
  if (tid < 16) {
    const int r = tid;
    float tv[TOPK]; int ti[TOPK];
#pragma unroll
    for (int e = 0; e < TOPK; ++e) { tv[e] = -__builtin_inff(); ti[e] = 0; }
    for (int ww = 0; ww < 4; ++ww) {
#pragma unroll
      for (int e = 0; e < TOPK; ++e) {
        float v = listV[ww][r][e];
        int   idx = listI[ww][r][e];
#pragma unroll
        for (int k = 0; k < TOPK; ++k) {
          if (v > tv[k]) {
            const float a = tv[k]; const int bi = ti[k];
            tv[k] = v; ti[k] = idx;
            v = a; idx = bi;
          }
        }
      }
    }
    const float m = tv[0];
    float ev[TOPK]; float s = 0.f;
#pragma unroll
    for (int k = 0; k < TOPK; ++k) { ev[k] = __expf(tv[k] - m); s += ev[k]; }
    const float inv = 1.0f / s;
    float* orow = out + ((size_t)b * NN + rtile + r) * NN;
#pragma unroll
    for (int k = 0; k < TOPK; ++k) orow[ti[k]] = ev[k] * inv;
  }
}

// ---------------------------------------------------------------------------
extern "C" void kernel_launch(void* const* d_in, const int* in_sizes, int n_in,
                              void* d_out, int out_size, void* d_ws, size_t ws_size,
                              hipStream_t stream) {
  const float* x  = (const float*)d_in[0];   // [B, N, C]
  const float* Wq = (const float*)d_in[1];   // [C, DK]
  const float* Wk = (const float*)d_in[2];   // [C, DK]
  // d_in[3] = top_k scalar (compile-time TOPK=8)

  float* Qws = (float*)d_ws;                           // [B*N, DK]
  float* Kws = Qws + (size_t)BB * NN * DK;             // [B*N, DK]

  // Output is mostly zeros: zero-fill (268 MB @ 23.3 TB/s ~ 11.5 us), then
  // scatter only B*N*8 softmax values.
  hipMemsetAsync(d_out, 0, (size_t)out_size * sizeof(float), stream);

  qk_proj_kernel<<<(BB * NN) / 16, 32, 0, stream>>>(x, Wq, Wk, Qws, Kws);
  topk_attn_kernel<<<(BB * NN) / 16, 128, 0, stream>>>(Qws, Kws, (float*)d_out);
}